// ProgramSynthesizer_87076166960171
// MI455X (gfx1250) — compile-verified
//
#include <hip/hip_runtime.h>
#include <math.h>

typedef __attribute__((ext_vector_type(2))) float v2f;
typedef __attribute__((ext_vector_type(8))) float v8f;

#define B_ 32
#define L_ 2048
#define D_ 1024
#define H_ 256
#define R_ 8
#define RD_ 8192
#define CI_ 9216
#define G3_ 768
#define NSTEPS 4
#define INJECT_ 0.1f
#define LCHUNKS 8

__device__ __forceinline__ float gelu_exact(float x) {
  return 0.5f * x * (1.0f + erff(x * 0.70710678118654752f));
}
__device__ __forceinline__ float sigmoidf_(float x) { return 1.0f / (1.0f + expf(-x)); }

__device__ __forceinline__ float block_reduce_sum(float v, float* red) {
  int t = threadIdx.x;
  red[t] = v;
  __syncthreads();
  for (int s = 128; s > 0; s >>= 1) {
    if (t < s) red[t] += red[t + s];
    __syncthreads();
  }
  float r = red[0];
  __syncthreads();
  return r;
}

// ---------------- pooled = x.mean(axis=1), stage 1: partial sums over L chunks ----
__global__ __launch_bounds__(256) void k_pool_part(const float* __restrict__ x,
                                                   float* __restrict__ ppool) {
  int b = blockIdx.y;
  int c = blockIdx.z;
  int d = blockIdx.x * 256 + threadIdx.x;
  const int lpc = L_ / LCHUNKS;
  const float* xp = x + (size_t)b * L_ * D_ + (size_t)c * lpc * D_ + d;
  float s = 0.f;
#pragma unroll 4
  for (int l = 0; l < lpc; ++l) s += xp[(size_t)l * D_];
  ppool[((size_t)b * LCHUNKS + c) * D_ + d] = s;
}

// ---------------- pooled stage 2: reduce chunks ----------------
__global__ __launch_bounds__(256) void k_pool_final(const float* __restrict__ ppool,
                                                    float* __restrict__ pooled) {
  int b = blockIdx.y;
  int d = blockIdx.x * 256 + threadIdx.x;
  float s = 0.f;
#pragma unroll
  for (int c = 0; c < LCHUNKS; ++c) s += ppool[((size_t)b * LCHUNKS + c) * D_ + d];
  pooled[b * D_ + d] = s * (1.0f / (float)L_);
}

// ---------------- init h, halt ----------------
__global__ __launch_bounds__(256) void k_init(const float* __restrict__ h0,
                                              float* __restrict__ h,
                                              float* __restrict__ halt) {
  int b = blockIdx.x, j = threadIdx.x;
  h[b * H_ + j] = h0[j];
  if (j == 0) halt[b] = 0.f;
}

// ---------------- ci = concat(pooled, regs) ----------------
__global__ __launch_bounds__(256) void k_pack_ci(const float* __restrict__ pooled,
                                                 const float* __restrict__ regs,
                                                 float* __restrict__ ci) {
  int b = blockIdx.y;
  int i = blockIdx.x * 256 + threadIdx.x;  // 0..9215
  float v = (i < D_) ? pooled[b * D_ + i] : regs[b * RD_ + (i - D_)];
  ci[b * CI_ + i] = v;
}

// ---------------- C[32,N] = A[32,K] @ W[K,N] via V_WMMA_F32_16X16X4_F32 ----------
// 4 waves/block, one 16-col tile per wave; grid.y = K-split (partials when >1).
// A frag (16x4 f32): lanes 0-15 -> M=lane, K={k,k+1}; lanes 16-31 -> M=lane-16, K={k+2,k+3}
// B frag (4x16 f32, mirrored): lane&15 = N col; half-wave+reg = K row.
// C/D (16x16 f32): vgpr r: lanes 0-15 -> M=r, lanes 16-31 -> M=r+8; N=lane&15
// Inner loop is pointer-increment form so addressing is v_add, not per-iter muls.
__global__ __launch_bounds__(128) void k_gemm32(const float* __restrict__ A,
                                                const float* __restrict__ W,
                                                const float* __restrict__ bias,
                                                float* __restrict__ C,
                                                int K, int N, int act, int Kc) {
  int lane = threadIdx.x & 31;
  int wave = threadIdx.x >> 5;
  int row = lane & 15;
  int koff = (lane >> 4) << 1;  // 0 or 2
  int n0 = (blockIdx.x * 4 + wave) * 16;
  int kbegin = blockIdx.y * Kc;
  const float* a0p = A + (size_t)row * K + kbegin + koff;
  const float* a1p = A + (size_t)(row + 16) * K + kbegin + koff;
  const float* wp = W + (size_t)(kbegin + koff) * N + n0 + row;
  const size_t wstep = (size_t)4 * N;
  v8f acc0 = {};
  v8f acc1 = {};
  int iters = Kc >> 2;
#pragma unroll 4
  for (int it = 0; it < iters; ++it) {
    v2f a0 = *(const v2f*)a0p;
    v2f a1 = *(const v2f*)a1p;
    v2f bb;
    bb[0] = wp[0];
    bb[1] = wp[N];
    a0p += 4;
    a1p += 4;
    wp += wstep;
    acc0 = __builtin_amdgcn_wmma_f32_16x16x4_f32(false, a0, false, bb, (short)0, acc0,
                                                 false, false);
    acc1 = __builtin_amdgcn_wmma_f32_16x16x4_f32(false, a1, false, bb, (short)0, acc1,
                                                 false, false);
  }
  int mb = (lane >> 4) * 8;
  if (gridDim.y == 1) {
    float bval = bias ? bias[n0 + row] : 0.0f;
#pragma unroll
    for (int rr = 0; rr < 8; ++rr) {
      float v0 = acc0[rr] + bval;
      float v1 = acc1[rr] + bval;
      if (act == 1) { v0 = gelu_exact(v0); v1 = gelu_exact(v1); }
      C[(size_t)(mb + rr) * N + n0 + row] = v0;
      C[(size_t)(16 + mb + rr) * N + n0 + row] = v1;
    }
  } else {
    float* Cp = C + (size_t)blockIdx.y * (B_ * (size_t)N);
#pragma unroll
    for (int rr = 0; rr < 8; ++rr) {
      Cp[(size_t)(mb + rr) * N + n0 + row] = acc0[rr];
      Cp[(size_t)(16 + mb + rr) * N + n0 + row] = acc1[rr];
    }
  }
}

// ---------------- sum K-split partials + bias + act ----------------
__global__ __launch_bounds__(256) void k_reduce(const float* __restrict__ part,
                                                const float* __restrict__ bias,
                                                float* __restrict__ C,
                                                int N, int ksplit, int act) {
  int i = blockIdx.x * 256 + threadIdx.x;  // over 32*N
  float s = 0.f;
  for (int k = 0; k < ksplit; ++k) s += part[(size_t)k * (B_ * (size_t)N) + i];
  if (bias) s += bias[i % N];
  if (act == 1) s = gelu_exact(s);
  C[i] = s;
}

// ---------------- GRU cell + softmaxes + register blends ----------------
__global__ __launch_bounds__(256) void k_gru(
    const float* __restrict__ gi, const float* __restrict__ gh,
    float* __restrict__ h, const float* __restrict__ regs,
    const float* __restrict__ u,
    const float* __restrict__ W_op, const float* __restrict__ b_op,
    const float* __restrict__ W_src, const float* __restrict__ b_src,
    const float* __restrict__ W_dst, const float* __restrict__ b_dst,
    const float* __restrict__ W_aux, const float* __restrict__ b_aux,
    float* __restrict__ opw, float* __restrict__ dstw,
    float* __restrict__ halt, float* __restrict__ live,
    float* __restrict__ src, float* __restrict__ dst, float* __restrict__ aux,
    float* __restrict__ sd, float* __restrict__ trace, int step) {
  __shared__ float sh_h[H_];
  __shared__ float sh_log[40];
  __shared__ float sh_opw[16], sh_sw[8], sh_dw[8], sh_aw[8];
  int b = blockIdx.x, t = threadIdx.x;
  const float* gib = gi + b * (3 * H_);
  const float* ghb = gh + b * (3 * H_);
  float hold = h[b * H_ + t];
  float r = sigmoidf_(gib[t] + ghb[t]);
  float z = sigmoidf_(gib[H_ + t] + ghb[H_ + t]);
  float n = tanhf(gib[2 * H_ + t] + r * ghb[2 * H_ + t]);
  float hn = (1.0f - z) * n + z * hold;
  h[b * H_ + t] = hn;
  sh_h[t] = hn;
  __syncthreads();
  if (t < 40) {
    float acc = 0.f;
    if (t < 16) {
      for (int k = 0; k < H_; ++k) acc += sh_h[k] * W_op[k * 16 + t];
      float uu = u[step * (B_ * 16) + b * 16 + t];
      acc += b_op[t] + (-logf(-logf(uu + 1e-20f) + 1e-20f));
    } else if (t < 24) {
      int o = t - 16;
      for (int k = 0; k < H_; ++k) acc += sh_h[k] * W_src[k * 8 + o];
      acc += b_src[o];
    } else if (t < 32) {
      int o = t - 24;
      for (int k = 0; k < H_; ++k) acc += sh_h[k] * W_dst[k * 8 + o];
      acc += b_dst[o];
    } else {
      int o = t - 32;
      for (int k = 0; k < H_; ++k) acc += sh_h[k] * W_aux[k * 8 + o];
      acc += b_aux[o];
    }
    sh_log[t] = acc;
  }
  __syncthreads();
  if (t == 0) {
    float m = sh_log[0];
    int am = 0;
    for (int o = 1; o < 16; ++o)
      if (sh_log[o] > m) { m = sh_log[o]; am = o; }
    float ssum = 0.f;
    for (int o = 0; o < 16; ++o) { float e = expf(sh_log[o] - m); sh_opw[o] = e; ssum += e; }
    float inv = 1.0f / ssum;
    for (int o = 0; o < 16; ++o) { sh_opw[o] *= inv; opw[b * 16 + o] = sh_opw[o]; }
    trace[b * NSTEPS + step] = (float)am;
    float hv = halt[b] + sh_opw[15];
    halt[b] = hv;
    live[b] = fmaxf(1.0f - hv, 0.0f);
  } else if (t >= 1 && t <= 3) {
    const int base = 16 + (t - 1) * 8;
    float m = sh_log[base];
    for (int o = 1; o < 8; ++o) m = fmaxf(m, sh_log[base + o]);
    float w[8];
    float ssum = 0.f;
    for (int o = 0; o < 8; ++o) { w[o] = expf(sh_log[base + o] - m); ssum += w[o]; }
    float inv = 1.0f / ssum;
    for (int o = 0; o < 8; ++o) {
      float v = w[o] * inv;
      if (t == 1) sh_sw[o] = v;
      else if (t == 2) { sh_dw[o] = v; dstw[b * 8 + o] = v; }
      else sh_aw[o] = v;
    }
  }
  __syncthreads();
  const float* rb = regs + b * RD_;
  for (int c = 0; c < 4; ++c) {
    int d = c * 256 + t;
    float sv = 0.f, dv = 0.f, av = 0.f;
#pragma unroll
    for (int rr = 0; rr < R_; ++rr) {
      float rv = rb[rr * D_ + d];
      sv += sh_sw[rr] * rv;
      dv += sh_dw[rr] * rv;
      av += sh_aw[rr] * rv;
    }
    src[b * D_ + d] = sv;
    dst[b * D_ + d] = dv;
    aux[b * D_ + d] = av;
    sd[b * D_ + d] = sv * dv;
  }
}

// ---------------- 16-op mixture + regs update ----------------
__global__ __launch_bounds__(256) void k_combine(
    const float* __restrict__ src, const float* __restrict__ dst,
    const float* __restrict__ attn, const float* __restrict__ proj,
    const float* __restrict__ outer,
    const float* __restrict__ opw, const float* __restrict__ dstw,
    const float* __restrict__ live, const float* __restrict__ op_scale,
    float* __restrict__ regs) {
  __shared__ float red[256];
  __shared__ float c_opw[16];
  __shared__ float c_dw[8];
  __shared__ float c_misc[2];
  int b = blockIdx.x, t = threadIdx.x;
  if (t < 16) c_opw[t] = opw[b * 16 + t];
  else if (t < 24) c_dw[t - 16] = dstw[b * 8 + (t - 16)];
  else if (t == 24) c_misc[0] = live[b];
  else if (t == 25) c_misc[1] = op_scale[0];
  __syncthreads();
  float sv[4];
  float sum = 0.f, sq = 0.f;
  for (int c = 0; c < 4; ++c) {
    float v = src[b * D_ + c * 256 + t];
    sv[c] = v;
    sum += v;
    sq += v * v;
  }
  sum = block_reduce_sum(sum, red);
  sq = block_reduce_sum(sq, red);
  float mean = sum * (1.0f / (float)D_);
  float var = sq * (1.0f / (float)D_) - mean * mean;
  float rstd = rsqrtf(var + 1e-5f);
  float lv = c_misc[0], oscale = c_misc[1];
  for (int c = 0; c < 4; ++c) {
    int d = c * 256 + t;
    float s = sv[c];
    float dd = dst[b * D_ + d];
    float at = attn[b * D_ + d];
    float pj = proj[b * D_ + d];
    float ou = outer[b * D_ + d];
    float ln = (s - mean) * rstd;
    float res = c_opw[0] * dd + c_opw[1] * (dd + s) + c_opw[2] * (sigmoidf_(s) * dd)
              + c_opw[3] * ln + c_opw[4] * pj + c_opw[5] * (s + dd)
              + c_opw[6] * at - c_opw[7] * s + c_opw[8] * (s * oscale)
              + c_opw[9] * s + c_opw[10] * fmaxf(s, dd) + c_opw[11] * fminf(s, dd)
              + c_opw[12] * fmaxf(s, 0.0f) + c_opw[13] * tanhf(s)
              + c_opw[14] * ou + c_opw[15] * dd;
    float rl = res * lv;
    float* rp = regs + b * RD_ + d;
#pragma unroll
    for (int rr = 0; rr < R_; ++rr) rp[rr * D_] += c_dw[rr] * rl;
  }
}

// ---------------- prog_out = rmsnorm(prog_out, norm_w) in place ----------------
__global__ __launch_bounds__(256) void k_rms_prog(float* __restrict__ prog,
                                                  const float* __restrict__ nw) {
  __shared__ float red[256];
  int b = blockIdx.x, t = threadIdx.x;
  float4 v = ((const float4*)(prog + (size_t)b * D_))[t];
  float sq = v.x * v.x + v.y * v.y + v.z * v.z + v.w * v.w;
  sq = block_reduce_sum(sq, red);
  float rs = rsqrtf(sq * (1.0f / (float)D_) + 1e-6f);
  float4 nv = ((const float4*)nw)[t];
  float4 o;
  o.x = v.x * rs * nv.x;
  o.y = v.y * rs * nv.y;
  o.z = v.z * rs * nv.z;
  o.w = v.w * rs * nv.w;
  ((float4*)(prog + (size_t)b * D_))[t] = o;
}

// ---------------- x_prog = rmsnorm(x + 0.1*prog_out, norm_w), float4 path --------
__global__ __launch_bounds__(256) void k_xprog(const float* __restrict__ x,
                                               const float* __restrict__ prog,
                                               const float* __restrict__ nw,
                                               float* __restrict__ y) {
  __shared__ float red[256];
  size_t row = blockIdx.x;   // b*L + l
  int b = (int)(row >> 11);  // L = 2048
  int t = threadIdx.x;
  float4 xv = ((const float4*)(x + row * D_))[t];
  float4 pv = ((const float4*)(prog + (size_t)b * D_))[t];
  float4 v;
  v.x = xv.x + INJECT_ * pv.x;
  v.y = xv.y + INJECT_ * pv.y;
  v.z = xv.z + INJECT_ * pv.z;
  v.w = xv.w + INJECT_ * pv.w;
  float sq = v.x * v.x + v.y * v.y + v.z * v.z + v.w * v.w;
  sq = block_reduce_sum(sq, red);
  float rs = rsqrtf(sq * (1.0f / (float)D_) + 1e-6f);
  float4 nv = ((const float4*)nw)[t];
  float4 o;
  o.x = v.x * rs * nv.x;
  o.y = v.y * rs * nv.y;
  o.z = v.z * rs * nv.z;
  o.w = v.w * rs * nv.w;
  ((float4*)(y + row * D_))[t] = o;
}

// ---------------- host-side GEMM dispatch helper ----------------
static void gemm32(const float* A, const float* W, const float* bias, float* C,
                   int K, int N, int act, int ksplit, float* part, hipStream_t stream) {
  int ntb = N / 64;  // 4 waves (4 n-tiles) per block
  if (ksplit <= 1) {
    k_gemm32<<<dim3(ntb, 1), dim3(128), 0, stream>>>(A, W, bias, C, K, N, act, K);
  } else {
    k_gemm32<<<dim3(ntb, ksplit), dim3(128), 0, stream>>>(A, W, nullptr, part, K, N, 0,
                                                          K / ksplit);
    k_reduce<<<dim3((B_ * N) / 256), dim3(256), 0, stream>>>(part, bias, C, N, ksplit, act);
  }
}

extern "C" void kernel_launch(void* const* d_in, const int* in_sizes, int n_in,
                              void* d_out, int out_size, void* d_ws, size_t ws_size,
                              hipStream_t stream) {
  (void)in_sizes; (void)n_in; (void)out_size; (void)ws_size;
  const float* x      = (const float*)d_in[0];
  const float* u      = (const float*)d_in[1];
  const float* W_e1   = (const float*)d_in[2];
  const float* b_e1   = (const float*)d_in[3];
  const float* W_e2   = (const float*)d_in[4];
  const float* b_e2   = (const float*)d_in[5];
  const float* W_ih   = (const float*)d_in[6];
  const float* b_ih   = (const float*)d_in[7];
  const float* W_hh   = (const float*)d_in[8];
  const float* b_hh   = (const float*)d_in[9];
  const float* h0     = (const float*)d_in[10];
  const float* W_op   = (const float*)d_in[11];
  const float* b_op   = (const float*)d_in[12];
  const float* W_src  = (const float*)d_in[13];
  const float* b_src  = (const float*)d_in[14];
  const float* W_dst  = (const float*)d_in[15];
  const float* b_dst  = (const float*)d_in[16];
  const float* W_aux  = (const float*)d_in[17];
  const float* b_aux  = (const float*)d_in[18];
  const float* W_proj = (const float*)d_in[19];
  const float* op_scale = (const float*)d_in[20];
  const float* Wv     = (const float*)d_in[21];
  const float* bv     = (const float*)d_in[22];
  const float* Wo     = (const float*)d_in[23];
  const float* bo     = (const float*)d_in[24];
  const float* W_d1   = (const float*)d_in[25];
  const float* b_d1   = (const float*)d_in[26];
  const float* W_d2   = (const float*)d_in[27];
  const float* b_d2   = (const float*)d_in[28];
  const float* norm_w = (const float*)d_in[29];

  float* out       = (float*)d_out;
  float* out_x     = out;
  float* out_regs  = out + (size_t)B_ * L_ * D_;
  float* out_prog  = out_regs + (size_t)B_ * RD_;
  float* out_trace = out_prog + (size_t)B_ * D_;

  float* ws     = (float*)d_ws;
  float* pooled = ws;                        // 32768
  float* ppool  = pooled + B_ * D_;          // 262144
  float* ci     = ppool + B_ * LCHUNKS * D_; // 294912
  float* e1     = ci + B_ * CI_;             // 8192
  float* h      = e1 + B_ * H_;              // 8192
  float* gi     = h + B_ * H_;               // 24576
  float* gh     = gi + B_ * G3_;             // 24576
  float* opw    = gh + B_ * G3_;             // 512
  float* dstw   = opw + B_ * 16;             // 256
  float* haltb  = dstw + B_ * R_;            // 32
  float* liveb  = haltb + B_;                // 32
  float* srcv   = liveb + B_;                // 32768
  float* dstv   = srcv + B_ * D_;
  float* auxv   = dstv + B_ * D_;
  float* sdv    = auxv + B_ * D_;
  float* tv     = sdv + B_ * D_;
  float* attnv  = tv + B_ * D_;
  float* projv  = attnv + B_ * D_;
  float* outerv = projv + B_ * D_;
  float* d1     = outerv + B_ * D_;          // 8192
  float* part   = d1 + B_ * H_;              // up to 8*32*768 = 196608

  dim3 blk256(256);

  k_pool_part<<<dim3(D_ / 256, B_, LCHUNKS), blk256, 0, stream>>>(x, ppool);
  k_pool_final<<<dim3(D_ / 256, B_), blk256, 0, stream>>>(ppool, pooled);
  k_init<<<dim3(B_), blk256, 0, stream>>>(h0, h, haltb);
  gemm32(pooled, W_e1, b_e1, e1, D_, H_, 1, 4, part, stream);
  gemm32(e1, W_e2, b_e2, out_regs, H_, RD_, 0, 1, part, stream);

  for (int s = 0; s < NSTEPS; ++s) {
    k_pack_ci<<<dim3(CI_ / 256, B_), blk256, 0, stream>>>(pooled, out_regs, ci);
    gemm32(ci, W_ih, b_ih, gi, CI_, G3_, 0, 8, part, stream);
    gemm32(h, W_hh, b_hh, gh, H_, G3_, 0, 1, part, stream);
    k_gru<<<dim3(B_), blk256, 0, stream>>>(gi, gh, h, out_regs, u,
        W_op, b_op, W_src, b_src, W_dst, b_dst, W_aux, b_aux,
        opw, dstw, haltb, liveb, srcv, dstv, auxv, sdv, out_trace, s);
    gemm32(auxv, Wv, bv, tv, D_, D_, 0, 4, part, stream);
    gemm32(tv, Wo, bo, attnv, D_, D_, 0, 4, part, stream);
    gemm32(srcv, W_proj, nullptr, projv, D_, D_, 0, 4, part, stream);
    gemm32(sdv, W_proj, nullptr, outerv, D_, D_, 0, 4, part, stream);
    k_combine<<<dim3(B_), blk256, 0, stream>>>(srcv, dstv, attnv, projv, outerv,
        opw, dstw, liveb, op_scale, out_regs);
  }

  gemm32(out_regs, W_d1, b_d1, d1, RD_, H_, 1, 16, part, stream);
  gemm32(d1, W_d2, b_d2, out_prog, H_, D_, 0, 1, part, stream);
  k_rms_prog<<<dim3(B_), blk256, 0, stream>>>(out_prog, norm_w);
  k_xprog<<<dim3(B_ * L_), blk256, 0, stream>>>(x, out_prog, norm_w, out_x);
}